// BackProjector_88424786690486
// MI455X (gfx1250) — compile-verified
//
#include <hip/hip_runtime.h>
#include <hip/hip_bf16.h>
#include <math.h>

// ---- problem constants (from reference) ----
#define BSZ   8
#define HH    480
#define WW    640
#define NPIX  (HH * WW)          // 307200
#define VR    100
#define NZ    80                  // MAXH - MINH = 72-(-8)
#define NVOX  (VR * VR * NZ)      // 800000
#define LMAP  480
#define GS    0.05f
#define MINH  (-8)
#define MIN_Z 13
#define MAX_Z 25
#define CXP   319.5f
#define CZP   239.5f
#define DEGC  57.29577951308232f

// ---- flat float offsets inside d_out (return-order concat) ----
#define O_COORDS 0                                   // BS*N*4 = 9830400
#define O_FEAT   (BSZ * NPIX * 4)                    // 9830400
#define O_MAPS   (O_FEAT + BSZ * NPIX * 8)           // 29491200
#define O_POSES  (O_MAPS + BSZ * 4 * LMAP * LMAP)    // 36864000
#define O_REP    (O_POSES + BSZ * 3)                 // 36864024

typedef __attribute__((ext_vector_type(2))) float v2f;
typedef __attribute__((ext_vector_type(8))) float v8f;

// ---------------------------------------------------------------------------
// Pose kernel: current_poses -> d_out, per-batch rot/trans params -> ws
// params layout per batch (16 floats): rot[9] row-major, trans_g[3], trans_m[3]
// ---------------------------------------------------------------------------
__global__ void bp_pose(const float* __restrict__ pose_obs,
                        const float* __restrict__ poses_last,
                        float* __restrict__ out,
                        float* __restrict__ params) {
    int b = threadIdx.x;
    if (b >= BSZ) return;
    float p0 = poses_last[b * 3 + 0], p1 = poses_last[b * 3 + 1], p2 = poses_last[b * 3 + 2];
    float r0 = pose_obs[b * 3 + 0], r1 = pose_obs[b * 3 + 1], r2 = pose_obs[b * 3 + 2];
    float th = p2 / DEGC;
    float x = p0 + r0 * cosf(th) - r1 * sinf(th);
    float y = p1 + r0 * sinf(th) + r1 * cosf(th);
    float t = p2 + r2 * DEGC;
    t = fmodf(t - 180.0f, 360.0f) + 180.0f;
    t = fmodf(t + 180.0f, 360.0f) - 180.0f;
    out[O_POSES + b * 3 + 0] = x;
    out[O_POSES + b * 3 + 1] = y;
    out[O_POSES + b * 3 + 2] = t;
    float st0 = x * 20.0f;               // *100/RES
    float st1 = y * 20.0f;
    float tr  = (90.0f - t) * 0.017453292519943295f;
    float c = cosf(tr), s = sinf(tr);
    float* pb = params + b * 16;
    pb[0] = c;  pb[1] = -s; pb[2] = 0.0f;      // rot row 0
    pb[3] = s;  pb[4] =  c; pb[5] = 0.0f;      // rot row 1
    pb[6] = 0;  pb[7] =  0; pb[8] = 1.0f;      // rot row 2
    pb[9]  = st1; pb[10] = st0; pb[11] = (float)(-MINH);        // trans_g (+8 folded)
    pb[12] = st1 * GS; pb[13] = st0 * GS; pb[14] = (float)(-MINH) * GS; // trans_m
    pb[15] = 0.0f;
}

// ---------------------------------------------------------------------------
// common per-point geometry
// ---------------------------------------------------------------------------
__device__ __forceinline__ void point_geom(const float* __restrict__ obs_b, int i,
                                           float& X, float& depth, float& Z,
                                           bool& valid, int& g0, int& g1, int& g2, int& vox) {
    const float inv_foc = tanf(39.5f * 0.017453292519943295f) / 320.0f; // tan(HFOV/2)/(W/2)
    depth = obs_b[3 * NPIX + i] * 0.01f;
    int u = i % WW, row = i / WW;
    X = ((float)u - CXP) * depth * inv_foc + 2.5f;                      // + VR*GS/2
    Z = ((float)(HH - 1 - row) - CZP) * depth * inv_foc + 0.88f;        // + CAM_H
    valid = (X > 0.0f) & (X < 5.0f) & (depth > 0.0f) & (depth < 5.0f) &
            (Z > (float)MINH * GS) & (Z < 72.0f * GS);
    g0 = (int)floorf(X * 20.0f);
    g1 = (int)floorf(depth * 20.0f);
    g2 = (int)floorf(Z * 20.0f);
    int c0 = min(max(g0, 0), VR - 1);
    int c1 = min(max(g1, 0), VR - 1);
    int c2 = min(max(g2 - MINH, 0), NZ - 1);
    vox = (c0 * VR + c1) * NZ + c2;
}

// ---------------------------------------------------------------------------
// Pass 1: segment-min -> first point index per voxel
// ---------------------------------------------------------------------------
__global__ __launch_bounds__(256) void bp_first(const float* __restrict__ obs,
                                                unsigned* __restrict__ first) {
    int gp = blockIdx.x * blockDim.x + threadIdx.x;   // 0 .. BS*N-1
    int b = gp / NPIX, i = gp % NPIX;
    float X, depth, Z; bool valid; int g0, g1, g2, vox;
    point_geom(obs + (size_t)b * 4 * NPIX, i, X, depth, Z, valid, g0, g1, g2, vox);
    if (valid) atomicMin(&first[(size_t)b * NVOX + vox], (unsigned)i);
}

// ---------------------------------------------------------------------------
// Pass 2: WMMA transforms + masked writes + map scatter.
// 256 threads = 8 waves, 16 points per wave (both lane-halves mirror a point).
// WMMA f32 16x16x4:  A lane<16:{x,y} lane>=16:{z,1};  B col n: rot[n][*]+trans[n]
// ---------------------------------------------------------------------------
__global__ __launch_bounds__(256) void bp_main(const float* __restrict__ obs,
                                               const unsigned* __restrict__ first,
                                               const float* __restrict__ params,
                                               float* __restrict__ out) {
    __shared__ float gciS[8][16][3];
    __shared__ float cfS[8][16][3];

    const int lane = threadIdx.x & 31;
    const int w    = threadIdx.x >> 5;
    const int q    = lane & 15;           // point-in-wave / B column index
    const int half = lane >> 4;
    const int gp   = blockIdx.x * 128 + w * 16 + q;   // global point id
    const int b    = gp / NPIX;
    const int i    = gp % NPIX;
    const float* obs_b = obs + (size_t)b * 4 * NPIX;

    float X, depth, Z; bool valid; int g0, g1, g2, vox;
    point_geom(obs_b, i, X, depth, Z, valid, g0, g1, g2, vox);
    const bool rep = valid && (first[(size_t)b * NVOX + vox] == (unsigned)i);

    // grid-coord path input: (g.y, g.x + X1-240, g.z); metric path: (depth, X-2.5, Z)
    float gcfx = (float)g1, gcfy = (float)g0 - 50.0f, gcfz = (float)g2;
    float cfx = depth, cfy = X - 2.5f, cfz = Z;

    v2f ag; ag[0] = half ? gcfz : gcfx;  ag[1] = half ? 1.0f : gcfy;
    v2f am; am[0] = half ? cfz  : cfx;   am[1] = half ? 1.0f : cfy;

    // B (4x16): B[k][n] = rot[n][k] (k<3), B[3][n] = trans[n]; columns n>=3 zero
    const float* pb = params + b * 16;
    const bool ncol = q < 3;
    const int nn = ncol ? q : 0;
    float r0 = pb[nn * 3 + 0], r1 = pb[nn * 3 + 1], r2 = pb[nn * 3 + 2];
    float tg = pb[9 + nn], tm = pb[12 + nn];
    v2f bg; bg[0] = ncol ? (half ? r2 : r0) : 0.0f;
            bg[1] = ncol ? (half ? tg : r1) : 0.0f;
    v2f bm; bm[0] = bg[0];
            bm[1] = ncol ? (half ? tm : r1) : 0.0f;

    v8f zero = {0.f, 0.f, 0.f, 0.f, 0.f, 0.f, 0.f, 0.f};
    // D = A x B + 0  (EXEC is all-ones here; no divergence above)
    v8f dg = __builtin_amdgcn_wmma_f32_16x16x4_f32(false, ag, false, bg, (short)0, zero, false, false);
    v8f dm = __builtin_amdgcn_wmma_f32_16x16x4_f32(false, am, false, bm, (short)0, zero, false, false);

    // D layout: VGPR v, lanes 0-15 -> M=v, N=lane; lanes 16-31 -> M=8+v, N=lane-16
    if (q < 3) {
#pragma unroll
        for (int v = 0; v < 8; ++v) {
            int p = half * 8 + v;
            gciS[w][p][q] = rintf(dg[v]);   // round-half-even == jnp.round
            cfS[w][p][q]  = dm[v];
        }
    }
    __syncthreads();

    if (lane < 16) {
        const float gx = gciS[w][q][0], gy = gciS[w][q][1], gz = gciS[w][q][2];
        const float repf = rep ? 1.0f : 0.0f;

        float* oc = out + O_COORDS + (size_t)gp * 4;
        oc[0] = repf * (float)b;
        oc[1] = repf * gx; oc[2] = repf * gy; oc[3] = repf * gz;

        float* of = out + O_FEAT + (size_t)gp * 8;
        of[0] = repf * cfS[w][q][0];
        of[1] = repf * cfS[w][q][1];
        of[2] = repf * cfS[w][q][2];
        of[3] = repf * obs_b[0 * NPIX + i];
        of[4] = repf * obs_b[1 * NPIX + i];
        of[5] = repf * obs_b[2 * NPIX + i];
        of[6] = 0.0f; of[7] = 0.0f;

        out[O_REP + gp] = repf;

        if (rep) {
            int xs = min(max((int)gx, 0), LMAP - 1);
            int ys = min(max((int)gy, 0), LMAP - 1);
            unsigned one = 0x3F800000u;   // 1.0f bits; int-max valid for non-neg floats
            unsigned* mb = (unsigned*)(out + O_MAPS) + (size_t)b * 4 * LMAP * LMAP;
            atomicMax(&mb[1 * LMAP * LMAP + xs * LMAP + ys], one);   // lm1
            int gzi = (int)gz;
            if (gzi >= MIN_Z && gzi <= MAX_Z)
                atomicMax(&mb[0 * LMAP * LMAP + xs * LMAP + ys], one); // lm0
        }
    }
}

// ---------------------------------------------------------------------------
extern "C" void kernel_launch(void* const* d_in, const int* in_sizes, int n_in,
                              void* d_out, int out_size, void* d_ws, size_t ws_size,
                              hipStream_t stream) {
    const float* obs        = (const float*)d_in[0];
    const float* pose_obs   = (const float*)d_in[1];
    // d_in[2] coords_last, d_in[3] feats_last: unused by reference outputs
    const float* poses_last = (const float*)d_in[4];
    const float* map_last   = (const float*)d_in[5];
    float* out = (float*)d_out;

    unsigned* first = (unsigned*)d_ws;                                   // BS*NVOX u32 (25.6 MB)
    float* params   = (float*)((char*)d_ws + (size_t)BSZ * NVOX * sizeof(unsigned));

    // segment-min init to 0xFFFFFFFF (unsigned infinity)
    hipMemsetAsync(first, 0xFF, (size_t)BSZ * NVOX * sizeof(unsigned), stream);
    // maps := map_last (channels 2/3 pass through; 0/1 get atomicMax'ed on top)
    hipMemcpyAsync(out + O_MAPS, map_last,
                   (size_t)BSZ * 4 * LMAP * LMAP * sizeof(float),
                   hipMemcpyDeviceToDevice, stream);

    bp_pose<<<1, 32, 0, stream>>>(pose_obs, poses_last, out, params);
    bp_first<<<(BSZ * NPIX) / 256, 256, 0, stream>>>(obs, first);
    bp_main<<<(BSZ * NPIX) / 128, 256, 0, stream>>>(obs, first, params, out);
}